// MotionDiscreteAE_34359738655
// MI455X (gfx1250) — compile-verified
//
#include <hip/hip_runtime.h>
#include <hip/hip_bf16.h>

typedef __attribute__((ext_vector_type(2))) float v2f;
typedef __attribute__((ext_vector_type(8))) float v8f;

#define TOKENS  65536     // B*N
#define KCODES  1024
#define DSLICE  256
#define MTILE   128       // tokens per workgroup (8 waves x 16 rows)
#define NTILE   16        // codes per column tile
#define NTILES  (KCODES / NTILE)   // 64

// ---------------------------------------------------------------------------
// Init: zero the ids+vq tail of d_out (atomics accumulate into it) and
// precompute per-code squared norms ||w||^2 into workspace.
// ---------------------------------------------------------------------------
__global__ __launch_bounds__(256) void vq_init_kernel(const float* __restrict__ W,
                                                      float* __restrict__ wsq,
                                                      float* __restrict__ out_tail,
                                                      int ntail) {
  const int t = blockIdx.x * blockDim.x + threadIdx.x;
  if (t < ntail) out_tail[t] = 0.0f;
  if (t < 2 * KCODES) {
    const float4* __restrict__ w4 = (const float4*)(W + (size_t)t * DSLICE);
    float ssum = 0.0f;
#pragma unroll 8
    for (int i = 0; i < DSLICE / 4; ++i) {
      const float4 v = w4[i];
      ssum += v.x * v.x + v.y * v.y + v.z * v.z + v.w * v.w;
    }
    wsq[t] = ssum;
  }
}

// ---------------------------------------------------------------------------
// Main: fused distance-GEMM (f32 WMMA) + argmin + gather + loss partials.
// Double-buffered B tiles filled with GLOBAL_LOAD_ASYNC_TO_LDS_B128 (ASYNCcnt),
// one barrier per tile, DMA overlapped with the WMMA chain.
// grid.x = 2 slices * 512 row-tiles; block = 256 threads = 8 waves.
// ---------------------------------------------------------------------------
__global__ __launch_bounds__(256) void vq_main_kernel(const float* __restrict__ h,
                                                      const float* __restrict__ W,
                                                      const float* __restrict__ wsq,
                                                      float* __restrict__ out) {
  __shared__ float Wt[2][NTILE][DSLICE + 4];  // +4 pad: conflict-free col reads
  __shared__ float redMin[8][16];
  __shared__ float redXsq[8][16];
  __shared__ int   redArg[8][16];

  const int s     = blockIdx.x >> 9;        // slice (512 row-tiles per slice)
  const int mbase = (blockIdx.x & 511) * MTILE;
  const int tid   = threadIdx.x;
  const int w     = tid >> 5;               // wave id 0..7
  const int lane  = tid & 31;
  const int col   = lane & 15;
  const int hi    = lane >> 4;              // half-wave select (K-pair 0/1 vs 2/3)

  const float* __restrict__ Wsl   = W + (size_t)s * KCODES * DSLICE;
  const float* __restrict__ wsq_s = wsq + s * KCODES;

  // This thread's 4 async-copy slots within a tile: element e = it*1024 + tid*4
  int cRow[4], cCol[4];
#pragma unroll
  for (int it = 0; it < 4; ++it) {
    const int e = (it << 10) + (tid << 2);
    cRow[it] = e >> 8;
    cCol[it] = e & 255;
  }

  // ---- preload A fragments: this wave's 16 rows x 256 dims, kept in VGPRs ----
  // 16x4 f32 A layout: lane m%16 holds K = 4*kk + {0,1} (lanes 0-15) / {2,3} (16-31)
  const int    row  = mbase + w * 16 + col;
  const float* hrow = h + (size_t)row * 512 + s * DSLICE;
  v2f   afrag[64];
  float xsq_half = 0.0f;
#pragma unroll
  for (int kk = 0; kk < 64; ++kk) {
    const v2f a = *(const v2f*)(hrow + (kk << 2) + (hi << 1));
    afrag[kk] = a;
    xsq_half += a.x * a.x + a.y * a.y;
  }
  const float xsq = xsq_half + __shfl_xor(xsq_half, 16, 32);  // full ||x_row||^2

  // ---- prologue: async-DMA tile 0 into buffer 0 ----
#pragma unroll
  for (int it = 0; it < 4; ++it) {
    const unsigned lds_off = (unsigned)(size_t)&Wt[0][cRow[it]][cCol[it]];
    const unsigned long long gaddr =
        (unsigned long long)(size_t)(Wsl + (size_t)cRow[it] * DSLICE + cCol[it]);
    asm volatile("global_load_async_to_lds_b128 %0, %1, off"
                 :: "v"(lds_off), "v"(gaddr) : "memory");
  }

  float runmin[8];
  int   runarg[8];
#pragma unroll
  for (int j = 0; j < 8; ++j) { runmin[j] = 3.0e38f; runarg[j] = 0; }

  // ---- sweep codebook in 16-code tiles: dist' = ||w||^2 - 2 x.w ----
  for (int t = 0; t < NTILES; ++t) {
    const int k0  = t << 4;
    const int cur = t & 1;

    asm volatile("s_wait_asynccnt 0x0" ::: "memory");  // own tile-t DMA done
    __syncthreads();  // tile t visible to all waves; buffer cur^1 free to refill

    if (t + 1 < NTILES) {  // overlap next tile's DMA with this tile's WMMAs
#pragma unroll
      for (int it = 0; it < 4; ++it) {
        const unsigned lds_off = (unsigned)(size_t)&Wt[cur ^ 1][cRow[it]][cCol[it]];
        const unsigned long long gaddr = (unsigned long long)(size_t)
            (Wsl + (size_t)(k0 + NTILE + cRow[it]) * DSLICE + cCol[it]);
        asm volatile("global_load_async_to_lds_b128 %0, %1, off"
                     :: "v"(lds_off), "v"(gaddr) : "memory");
      }
    }
    const float wsq_l = wsq_s[k0 + col];

    // Two independent accumulator chains to avoid per-WMMA D->C delay slots.
    v8f acc0 = {}, acc1 = {};
#pragma unroll
    for (int kk = 0; kk < 64; kk += 2) {
      const v2f b0 = *(const v2f*)(&Wt[cur][col][(kk << 2) + (hi << 1)]);
      const v2f b1 = *(const v2f*)(&Wt[cur][col][((kk + 1) << 2) + (hi << 1)]);
      acc0 = __builtin_amdgcn_wmma_f32_16x16x4_f32(false, afrag[kk], false, b0,
                                                   (short)0, acc0, false, false);
      acc1 = __builtin_amdgcn_wmma_f32_16x16x4_f32(false, afrag[kk + 1], false, b1,
                                                   (short)0, acc1, false, false);
    }
#pragma unroll
    for (int j = 0; j < 8; ++j) {           // entry j: rows j / j+8, code k0+col
      const float d = wsq_l - 2.0f * (acc0[j] + acc1[j]);
      if (d < runmin[j]) { runmin[j] = d; runarg[j] = k0 + col; }
    }
  }

  // ---- argmin across the 16 lanes of each half (tie -> lowest code id) ----
#pragma unroll
  for (int off = 1; off < 16; off <<= 1) {
#pragma unroll
    for (int j = 0; j < 8; ++j) {
      const float ov = __shfl_xor(runmin[j], off, 32);
      const int   oa = __shfl_xor(runarg[j], off, 32);
      if (ov < runmin[j] || (ov == runmin[j] && oa < runarg[j])) {
        runmin[j] = ov; runarg[j] = oa;
      }
    }
  }
  if (lane == 0) {
#pragma unroll
    for (int j = 0; j < 8; ++j) { redMin[w][j] = runmin[j]; redArg[w][j] = runarg[j]; }
  }
  if (lane == 16) {
#pragma unroll
    for (int j = 0; j < 8; ++j) { redMin[w][8 + j] = runmin[j]; redArg[w][8 + j] = runarg[j]; }
  }
  if (lane < 16) redXsq[w][lane] = xsq;
  __syncthreads();

  float* __restrict__ zout = out;                               // [65536, 512]
  float* __restrict__ ids  = out + (size_t)TOKENS * 512;        // [65536]
  float* __restrict__ vqp  = ids + TOKENS;                      // [1]

  // vq_total partial: 1.25 * sum(min dist) / (BN*DS); min dist = ||x||^2 + dist'
  float vq = (lane < 16) ? (redMin[w][lane] + redXsq[w][lane]) : 0.0f;
#pragma unroll
  for (int off = 16; off >= 1; off >>= 1) vq += __shfl_xor(vq, off, 32);
  if (lane == 0) atomicAdd(vqp, vq * (1.25f / (65536.0f * 256.0f)));

  // packed ids: id0 + 1024*id1 (exact in f32, order-independent)
  if (lane < 16) {
    const int tok = mbase + w * 16 + lane;
    atomicAdd(ids + tok, (float)(redArg[w][lane] * (s ? KCODES : 1)));
  }

  // z = gathered codebook rows (straight-through forward value == z_q)
#pragma unroll 4
  for (int r = 0; r < 16; ++r) {
    const int arg = redArg[w][r];
    const float4* __restrict__ src = (const float4*)(Wsl + (size_t)arg * DSLICE);
    float4* __restrict__ dst =
        (float4*)(zout + (size_t)(mbase + w * 16 + r) * 512 + s * DSLICE);
    dst[lane]      = src[lane];        // 128 floats
    dst[lane + 32] = src[lane + 32];   // 128 floats
  }
}

// ---------------------------------------------------------------------------
extern "C" void kernel_launch(void* const* d_in, const int* in_sizes, int n_in,
                              void* d_out, int out_size, void* d_ws, size_t ws_size,
                              hipStream_t stream) {
  const float* h = (const float*)d_in[0];   // [32, 2048, 512]
  const float* W = (const float*)d_in[1];   // [2, 1024, 256]
  float* out = (float*)d_out;               // z | ids | vq_total
  float* wsq = (float*)d_ws;                // [2*1024] code norms

  float* out_tail = out + (size_t)TOKENS * 512;
  const int ntail = TOKENS + 1;             // ids + vq slot (atomically built)

  vq_init_kernel<<<(ntail + 255) / 256, 256, 0, stream>>>(W, wsq, out_tail, ntail);
  vq_main_kernel<<<2 * (TOKENS / MTILE), 256, 0, stream>>>(h, W, wsq, out);
}